// RandomLowRes3D_76003741270471
// MI455X (gfx1250) — compile-verified
//
#include <hip/hip_runtime.h>

// RandomLowRes3D for MI455X (gfx1250, wave32).
// Pipeline per (b,c) volume of 160^3 f32:
//   conv_w (WMMA) -> conv_h (WMMA) -> conv_d (WMMA) -> trilinear down -> trilinear up
// Separable Gaussian conv is cast as a banded-Toeplitz matmul on
// V_WMMA_F32_16X16X4_F32 (16x16 output tile per wave, K = L+15 im2col).
// Toeplitz operand reads come from a zero-padded LDS tap table so the inner
// loop is branchless (no EXEC-predicated ds_loads next to the WMMA).
// Memory-bound workload (~1.2 GB traffic @ 23.3 TB/s ~ 50us); intermediates
// largely live in the 192 MB L2.

#define DIM   160
#define PLANE (DIM * DIM)          // 25600
#define VOL   (DIM * DIM * DIM)    // 4096000
#define NVOL  8                    // B*C = 2*4
#define PSTR  32                   // params stride (floats) per volume
#define TAPPAD 16                  // zero padding before taps in LDS table
// params layout per volume: [0..23]=kernel taps, [24]=radius, [25]=down, [26]=res

typedef __attribute__((ext_vector_type(2))) float v2f;
typedef __attribute__((ext_vector_type(8))) float v8f;

__device__ __forceinline__ int reflect160(int i) {
    // 'symmetric' (reflect including edge); single reflection suffices here.
    if (i < 0)    i = -i - 1;
    if (i >= DIM) i = 2 * DIM - 1 - i;
    return i;
}

// ---------------------------------------------------------------- params ----
__global__ void lr_params(const float* __restrict__ resIn, float* __restrict__ params) {
    int v = threadIdx.x;
    if (v >= NVOL) return;
    float res   = fmaxf(resIn[v], 1.0f);
    float sigma = res * 0.42466090014400953f;           // FWHM_TO_SIGMA
    int   radius = (int)ceilf(3.0f * sigma);
    if (radius < 1) radius = 1;
    float denom = sigma * 1.4142135623730951f;
    float k[25];
    float sum = 0.0f;
    int L = 2 * radius + 1;
    for (int i = -radius; i <= radius; ++i) {
        float val = 0.5f * (erff(((float)i + 0.5f) / denom) -
                            erff(((float)i - 0.5f) / denom));
        k[i + radius] = val;
        sum += val;
    }
    float inv = 1.0f / sum;
    float* p = params + v * PSTR;
    for (int i = 0; i < 24; ++i) p[i] = (i < L) ? k[i] * inv : 0.0f;
    p[24] = (float)radius;
    int down = (int)floorf((float)DIM / res);
    if (down < 1) down = 1;
    p[25] = (float)down;
    p[26] = res;
}

// Build zero-padded tap table in LDS: tap[TAPPAD + u] = k[u] for u in [0,L),
// zero elsewhere. Valid lookup range u in [-15, 40] -> index in [1, 56].
__device__ __forceinline__ void load_taps(float* tap, const float* params,
                                          int vol, int L) {
    if (threadIdx.x < 64) {
        int u = (int)threadIdx.x - TAPPAD;
        tap[threadIdx.x] = (u >= 0 && u < L) ? params[vol * PSTR + u] : 0.0f;
    }
    __syncthreads();
}

// ------------------------------------------------- conv along W (contig) ----
// Wave tile: D(16 lines x 16 positions) = A(data 16xK) * B(Toeplitz Kx16).
// Stores are coalesced along W (N = lane = position).
__global__ void lr_conv_w(const float* __restrict__ src, float* __restrict__ dst,
                          const float* __restrict__ params) {
    const int wave = threadIdx.x >> 5;
    const int lane = threadIdx.x & 31;
    const int bid  = blockIdx.x;
    const int vol  = bid / 4000;                 // 4 tiles/block, 16000 tiles/vol
    const int tile = (bid % 4000) * 4 + wave;
    const int lineTile = tile / 10;              // 0..1599  (25600/16 lines)
    const int posTile  = tile % 10;              // 0..9

    const int radius = (int)params[vol * PSTR + 24];
    const int L = 2 * radius + 1;
    const int K = L + 15;

    __shared__ float tap[64];
    load_taps(tap, params, vol, L);

    const float* s = src + (size_t)vol * VOL;
    float*       d = dst + (size_t)vol * VOL;

    const int m       = lane & 15;
    const int koff    = (lane < 16) ? 0 : 2;
    const int lineBase = lineTile * 16;
    const int outBase  = posTile * 16;
    const float* linep = s + (size_t)(lineBase + m) * DIM;   // A operand line
    const float* bandp = tap + TAPPAD - m;                   // branchless band

    v8f acc = {};
    for (int c4 = 0; c4 < K; c4 += 4) {
        // A = data (im2col), lanes 0-15: K pair (0,1), lanes 16-31: (2,3)
        int i0 = outBase - radius + c4 + koff;
        v2f a;
        a.x = linep[reflect160(i0)];
        a.y = linep[reflect160(i0 + 1)];
        // B = Toeplitz band, N = lane&15 = output position (consecutive LDS reads)
        int kk = c4 + koff;
        v2f b;
        b.x = bandp[kk];
        b.y = bandp[kk + 1];
        acc = __builtin_amdgcn_wmma_f32_16x16x4_f32(false, a, false, b,
                                                    (short)0, acc, false, false);
    }
    // C/D layout: VGPR v -> row M = v (+8 for upper half), col N = lane&15
    const int half8 = (lane < 16) ? 0 : 8;
    #pragma unroll
    for (int v = 0; v < 8; ++v) {
        d[(size_t)(lineBase + v + half8) * DIM + outBase + m] = acc[v];
    }
}

// -------------------------------------- conv along a strided axis (H, D) ----
// Lanes map to 16 consecutive W (coalesced loads & stores); rows = positions.
// H-conv: strideO=PLANE (o=d), strideC=DIM.  D-conv: strideO=DIM (o=h), strideC=PLANE.
__global__ void lr_conv_s(const float* __restrict__ src, float* __restrict__ dst,
                          const float* __restrict__ params, int strideO, int strideC) {
    const int wave = threadIdx.x >> 5;
    const int lane = threadIdx.x & 31;
    const int bid  = blockIdx.x;
    const int vol  = bid / 4000;
    const int tile = (bid % 4000) * 4 + wave;     // 0..15999 = o(160)*wt(10)*pt(10)
    const int o  = tile / 100;
    const int wt = (tile / 10) % 10;
    const int pt = tile % 10;

    const int radius = (int)params[vol * PSTR + 24];
    const int L = 2 * radius + 1;
    const int K = L + 15;

    __shared__ float tap[64];
    load_taps(tap, params, vol, L);

    const float* s = src + (size_t)vol * VOL + (size_t)o * strideO;
    float*       d = dst + (size_t)vol * VOL + (size_t)o * strideO;

    const int mm      = lane & 15;
    const int koff    = (lane < 16) ? 0 : 2;
    const int wBase   = wt * 16;
    const int outBase = pt * 16;
    const int w = wBase + mm;
    const float* bandp = tap + TAPPAD - mm;

    v8f acc = {};
    for (int c4 = 0; c4 < K; c4 += 4) {
        int kk = c4 + koff;
        // A = Toeplitz band, rows M = output positions (consecutive LDS reads)
        v2f a;
        a.x = bandp[kk];
        a.y = bandp[kk + 1];
        // B = data, N = lane&15 -> consecutive W (coalesced)
        int p0 = outBase - radius + kk;
        v2f b;
        b.x = s[(size_t)reflect160(p0)     * strideC + w];
        b.y = s[(size_t)reflect160(p0 + 1) * strideC + w];
        acc = __builtin_amdgcn_wmma_f32_16x16x4_f32(false, a, false, b,
                                                    (short)0, acc, false, false);
    }
    const int half8 = (lane < 16) ? 0 : 8;
    #pragma unroll
    for (int v = 0; v < 8; ++v) {
        d[(size_t)(outBase + v + half8) * strideC + w] = acc[v];
    }
}

// ------------------------------------------------------ trilinear resample ----
// Downsample: smoothed 160^3 -> down^3 (packed, x fastest), coords = i*res.
__global__ void lr_down(const float* __restrict__ src, float* __restrict__ dst,
                        const float* __restrict__ params) {
    const int vol = blockIdx.x / 16000;
    const int idx = (blockIdx.x % 16000) * 256 + threadIdx.x;
    const int down = (int)params[vol * PSTR + 25];
    const float res = params[vol * PSTR + 26];
    const int n = down * down * down;
    if (idx >= n) return;

    const int x = idx % down;
    const int y = (idx / down) % down;
    const int z = idx / (down * down);

    float cz = fminf((float)z * res, (float)(DIM - 1));
    float cy = fminf((float)y * res, (float)(DIM - 1));
    float cx = fminf((float)x * res, (float)(DIM - 1));
    int z0 = min((int)floorf(cz), DIM - 2);
    int y0 = min((int)floorf(cy), DIM - 2);
    int x0 = min((int)floorf(cx), DIM - 2);
    float wz = cz - (float)z0, wy = cy - (float)y0, wx = cx - (float)x0;

    const float* s = src + (size_t)vol * VOL;
    float v = 0.0f;
    #pragma unroll
    for (int dz = 0; dz < 2; ++dz) {
        float fz = dz ? wz : 1.0f - wz;
        #pragma unroll
        for (int dy = 0; dy < 2; ++dy) {
            float fy = dy ? wy : 1.0f - wy;
            #pragma unroll
            for (int dx = 0; dx < 2; ++dx) {
                float fx = dx ? wx : 1.0f - wx;
                v += fz * fy * fx *
                     s[(size_t)(z0 + dz) * PLANE + (size_t)(y0 + dy) * DIM + (x0 + dx)];
            }
        }
    }
    dst[(size_t)vol * VOL + idx] = v;
}

// Upsample: down^3 (packed) -> 160^3, coords = i/res clipped to [0, down-1].
__global__ void lr_up(const float* __restrict__ src, float* __restrict__ dst,
                      const float* __restrict__ params) {
    const int vol = blockIdx.x / 16000;
    const int idx = (blockIdx.x % 16000) * 256 + threadIdx.x;
    const int down = (int)params[vol * PSTR + 25];
    const float res = params[vol * PSTR + 26];

    const int x = idx % DIM;
    const int y = (idx / DIM) % DIM;
    const int z = idx / PLANE;

    float inv = 1.0f / res;
    float lim = (float)(down - 1);
    float cz = fminf((float)z * inv, lim);
    float cy = fminf((float)y * inv, lim);
    float cx = fminf((float)x * inv, lim);
    int hi = max(down - 2, 0);
    int z0 = min((int)floorf(cz), hi);
    int y0 = min((int)floorf(cy), hi);
    int x0 = min((int)floorf(cx), hi);
    float wz = cz - (float)z0, wy = cy - (float)y0, wx = cx - (float)x0;

    const float* s = src + (size_t)vol * VOL;
    const int dd = down * down;
    float v = 0.0f;
    #pragma unroll
    for (int dz = 0; dz < 2; ++dz) {
        int zi = min(z0 + dz, down - 1);
        float fz = dz ? wz : 1.0f - wz;
        #pragma unroll
        for (int dy = 0; dy < 2; ++dy) {
            int yi = min(y0 + dy, down - 1);
            float fy = dy ? wy : 1.0f - wy;
            #pragma unroll
            for (int dx = 0; dx < 2; ++dx) {
                int xi = min(x0 + dx, down - 1);
                float fx = dx ? wx : 1.0f - wx;
                v += fz * fy * fx * s[(size_t)zi * dd + (size_t)yi * down + xi];
            }
        }
    }
    dst[(size_t)vol * VOL + idx] = v;
}

// ------------------------------------------------------------------ launch ----
extern "C" void kernel_launch(void* const* d_in, const int* in_sizes, int n_in,
                              void* d_out, int out_size, void* d_ws, size_t ws_size,
                              hipStream_t stream) {
    const float* x     = (const float*)d_in[0];
    const float* resol = (const float*)d_in[1];
    float* out = (float*)d_out;

    float* params = (float*)d_ws;                 // NVOL*PSTR floats
    float* bufA   = params + 64;                  // 131 MB scratch volume set

    lr_params<<<1, 32, 0, stream>>>(resol, params);

    const int convBlocks = NVOL * 4000;           // 4 wave-tiles per 128-thread block
    // x -> out (conv W), out -> bufA (conv H), bufA -> out (conv D)
    lr_conv_w<<<convBlocks, 128, 0, stream>>>(x, out, params);
    lr_conv_s<<<convBlocks, 128, 0, stream>>>(out, bufA, params, PLANE, DIM); // H
    lr_conv_s<<<convBlocks, 128, 0, stream>>>(bufA, out, params, DIM, PLANE); // D
    // out -> bufA (downsample), bufA -> out (upsample)
    const int rsBlocks = NVOL * 16000;            // 256 thr, covers 160^3 worst case
    lr_down<<<rsBlocks, 256, 0, stream>>>(out, bufA, params);
    lr_up<<<rsBlocks, 256, 0, stream>>>(bufA, out, params);
}